// DropLearner2_28200755266073
// MI455X (gfx1250) — compile-verified
//
#include <hip/hip_runtime.h>

#define N_NODES   50000
#define DIM       256
#define HID       64
#define N_LO_CNT  40000
#define TEMP_C    0.5f
#define BIAS_C    0.0001f

typedef __attribute__((ext_vector_type(2))) float v2f;
typedef __attribute__((ext_vector_type(8))) float v8f;

// ---------------------------------------------------------------------------
// Kernel 1: fused dual IIR filter (direct-form II transposed) along features.
// One thread per node row; both filters share the single read of node_emb.
// float4 (b128) loads/stores for full-width VMEM transactions.
// ---------------------------------------------------------------------------
__global__ void __launch_bounds__(256)
iir_kernel(const float* __restrict__ x,
           const float* __restrict__ bh_g, const float* __restrict__ ah_g,
           const float* __restrict__ bl_g, const float* __restrict__ al_g,
           float* __restrict__ hi, float* __restrict__ lo)
{
    int row = blockIdx.x * blockDim.x + threadIdx.x;
    if (row >= N_NODES) return;

    float bh[6], ah[5], bl[6], al[5];
    float ih = 1.0f / ah_g[0], il = 1.0f / al_g[0];
    #pragma unroll
    for (int i = 0; i < 6; ++i) { bh[i] = bh_g[i] * ih; bl[i] = bl_g[i] * il; }
    #pragma unroll
    for (int i = 0; i < 5; ++i) { ah[i] = ah_g[i + 1] * ih; al[i] = al_g[i + 1] * il; }

    float zh[5] = {0.f,0.f,0.f,0.f,0.f};
    float zl[5] = {0.f,0.f,0.f,0.f,0.f};

    const float4* xr = (const float4*)(x  + (size_t)row * DIM);
    float4*       hr = (float4*)(hi + (size_t)row * DIM);
    float4*       lr = (float4*)(lo + (size_t)row * DIM);

    for (int q = 0; q < DIM / 4; ++q) {
        float4 xv = xr[q];
        float xin[4] = { xv.x, xv.y, xv.z, xv.w };
        float yh4[4], yl4[4];
        #pragma unroll
        for (int t = 0; t < 4; ++t) {
            float xn = xin[t];
            // high-pass
            float yh = fmaf(bh[0], xn, zh[0]);
            zh[0] = fmaf(bh[1], xn, zh[1]) - ah[0] * yh;
            zh[1] = fmaf(bh[2], xn, zh[2]) - ah[1] * yh;
            zh[2] = fmaf(bh[3], xn, zh[3]) - ah[2] * yh;
            zh[3] = fmaf(bh[4], xn, zh[4]) - ah[3] * yh;
            zh[4] = bh[5] * xn - ah[4] * yh;
            yh4[t] = yh;
            // low-pass
            float yl = fmaf(bl[0], xn, zl[0]);
            zl[0] = fmaf(bl[1], xn, zl[1]) - al[0] * yl;
            zl[1] = fmaf(bl[2], xn, zl[2]) - al[1] * yl;
            zl[2] = fmaf(bl[3], xn, zl[3]) - al[2] * yl;
            zl[3] = fmaf(bl[4], xn, zl[4]) - al[3] * yl;
            zl[4] = bl[5] * xn - al[4] * yl;
            yl4[t] = yl;
        }
        hr[q] = make_float4(yh4[0], yh4[1], yh4[2], yh4[3]);
        lr[q] = make_float4(yl4[0], yl4[1], yl4[2], yl4[3]);
    }
}

// ---------------------------------------------------------------------------
// Kernel 2: fused gather + MLP via V_WMMA_F32_16X16X4_F32 (fp32-precise).
// One wave (32 lanes) per 16-row tile; 4 accumulators cover H=64; K=256 in
// 64 steps of 4. N_LO (40000) is a multiple of 16 so tiles are uniform.
// ---------------------------------------------------------------------------
__global__ void __launch_bounds__(32)
mlp_wmma_kernel(const float* __restrict__ lo, const float* __restrict__ hi,
                const int* __restrict__ idx_lo, const int* __restrict__ idx_hi,
                const float* __restrict__ w1, const float* __restrict__ b1,
                const float* __restrict__ w2, const float* __restrict__ b2,
                float* __restrict__ wout)
{
    const int m0   = blockIdx.x * 16;
    const int lane = threadIdx.x;
    const int ln   = lane & 15;
    const int half = lane >> 4;          // 0: K pair {0,1}, 1: K pair {2,3}

    // Gathered source row for the A fragment (reconverges before WMMA).
    const int m = m0 + ln;
    const float* arow;
    if (m < N_LO_CNT) arow = lo + (size_t)idx_lo[m] * DIM;
    else              arow = hi + (size_t)idx_hi[m - N_LO_CNT] * DIM;
    const float* aptr = arow + half * 2;
    const float* bcol = w1 + (size_t)(half * 2) * HID + ln;   // W1[k0+2*half][ln + 16*t]

    v8f c0 = {}, c1 = {}, c2 = {}, c3 = {};

    #pragma unroll 4
    for (int k0 = 0; k0 < DIM; k0 += 4) {
        float2 av = *(const float2*)(aptr + k0);              // A[m][k0+2h .. k0+2h+1]
        v2f a; a.x = av.x; a.y = av.y;

        const float* bk = bcol + (size_t)k0 * HID;
        v2f b0v, b1v, b2v, b3v;
        b0v.x = bk[0];   b0v.y = bk[HID];                      // N-tile 0
        b1v.x = bk[16];  b1v.y = bk[HID + 16];                 // N-tile 1
        b2v.x = bk[32];  b2v.y = bk[HID + 32];                 // N-tile 2
        b3v.x = bk[48];  b3v.y = bk[HID + 48];                 // N-tile 3

        c0 = __builtin_amdgcn_wmma_f32_16x16x4_f32(false, a, false, b0v, (short)0, c0, false, false);
        c1 = __builtin_amdgcn_wmma_f32_16x16x4_f32(false, a, false, b1v, (short)0, c1, false, false);
        c2 = __builtin_amdgcn_wmma_f32_16x16x4_f32(false, a, false, b2v, (short)0, c2, false, false);
        c3 = __builtin_amdgcn_wmma_f32_16x16x4_f32(false, a, false, b3v, (short)0, c3, false, false);
    }

    // Epilogue: bias + ReLU + dot with w2, reduce 16 lanes per half-wave.
    // C layout: VGPR j <-> M = j + 8*half, N = ln (within tile).
    float s[8];
    #pragma unroll
    for (int j = 0; j < 8; ++j) s[j] = 0.f;

    float bb[4] = { b1[ln], b1[16 + ln], b1[32 + ln], b1[48 + ln] };
    float ww[4] = { w2[ln], w2[16 + ln], w2[32 + ln], w2[48 + ln] };
    v8f cc[4] = { c0, c1, c2, c3 };
    #pragma unroll
    for (int t = 0; t < 4; ++t) {
        #pragma unroll
        for (int j = 0; j < 8; ++j) {
            float h = cc[t][j] + bb[t];
            h = fmaxf(h, 0.f);
            s[j] = fmaf(h, ww[t], s[j]);
        }
    }

    const float bias2 = b2[0];
    #pragma unroll
    for (int j = 0; j < 8; ++j) {
        float v = s[j];
        v += __shfl_xor(v, 1, 32);
        v += __shfl_xor(v, 2, 32);
        v += __shfl_xor(v, 4, 32);
        v += __shfl_xor(v, 8, 32);   // masks < 16: reduction stays within each half
        if (ln == 0) wout[m0 + half * 8 + j] = v + bias2;
    }
}

// ---------------------------------------------------------------------------
// Kernel 3: edge gate + per-block partial sums (deterministic reduction tree).
// ---------------------------------------------------------------------------
__global__ void __launch_bounds__(256)
edge_kernel(const float* __restrict__ wsrc, const float* __restrict__ wdst,
            const int* __restrict__ src, const int* __restrict__ dst,
            const float* __restrict__ ueps, float* __restrict__ gate_out,
            float* __restrict__ partial, int E)
{
    const float invT = 1.0f / TEMP_C;
    float local = 0.f;
    for (int e = blockIdx.x * blockDim.x + threadIdx.x; e < E;
         e += gridDim.x * blockDim.x) {
        float w   = wsrc[src[e]] + wdst[dst[e]];
        float u   = ueps[e];
        float eps = (1.f - BIAS_C) - (1.f - 2.f * BIAS_C) * u;
        float z   = (logf(eps) - log1pf(-eps) + w) * invT;
        float g   = 1.f / (1.f + expf(-z));
        gate_out[e] = g;
        local += 1.f - g;
    }
    // wave32 reduction
    local += __shfl_xor(local, 16, 32);
    local += __shfl_xor(local,  8, 32);
    local += __shfl_xor(local,  4, 32);
    local += __shfl_xor(local,  2, 32);
    local += __shfl_xor(local,  1, 32);
    __shared__ float sdata[8];
    int wid = threadIdx.x >> 5, lid = threadIdx.x & 31;
    if (lid == 0) sdata[wid] = local;
    __syncthreads();
    if (threadIdx.x == 0) {
        float v = 0.f;
        #pragma unroll
        for (int i = 0; i < 8; ++i) v += sdata[i];
        partial[blockIdx.x] = v;
    }
}

__global__ void __launch_bounds__(256)
finalize_kernel(const float* __restrict__ partial, int nparts,
                float* __restrict__ out, float invE)
{
    float local = 0.f;
    for (int i = threadIdx.x; i < nparts; i += 256) local += partial[i];
    local += __shfl_xor(local, 16, 32);
    local += __shfl_xor(local,  8, 32);
    local += __shfl_xor(local,  4, 32);
    local += __shfl_xor(local,  2, 32);
    local += __shfl_xor(local,  1, 32);
    __shared__ float sdata[8];
    int wid = threadIdx.x >> 5, lid = threadIdx.x & 31;
    if (lid == 0) sdata[wid] = local;
    __syncthreads();
    if (threadIdx.x == 0) {
        float v = 0.f;
        #pragma unroll
        for (int i = 0; i < 8; ++i) v += sdata[i];
        out[0] = v * invE;
    }
}

// ---------------------------------------------------------------------------
extern "C" void kernel_launch(void* const* d_in, const int* in_sizes, int n_in,
                              void* d_out, int out_size, void* d_ws, size_t ws_size,
                              hipStream_t stream)
{
    const float* node_emb   = (const float*)d_in[0];
    const float* w1_src     = (const float*)d_in[1];
    const float* b1_src     = (const float*)d_in[2];
    const float* w2_src     = (const float*)d_in[3];
    const float* b2_src     = (const float*)d_in[4];
    const float* w1_dst     = (const float*)d_in[5];
    const float* b1_dst     = (const float*)d_in[6];
    const float* w2_dst     = (const float*)d_in[7];
    const float* b2_dst     = (const float*)d_in[8];
    const float* b_hi       = (const float*)d_in[9];
    const float* a_hi       = (const float*)d_in[10];
    const float* b_lo       = (const float*)d_in[11];
    const float* a_lo       = (const float*)d_in[12];
    const float* u_eps      = (const float*)d_in[13];
    const int*   src        = (const int*)d_in[14];
    const int*   dst        = (const int*)d_in[15];
    const int*   idx_src_hi = (const int*)d_in[16];
    const int*   idx_src_lo = (const int*)d_in[17];
    const int*   idx_dst_hi = (const int*)d_in[18];
    const int*   idx_dst_lo = (const int*)d_in[19];

    const int E = in_sizes[13];

    // Workspace layout (all offsets 256B-aligned)
    char*  ws      = (char*)d_ws;
    const size_t FILT = (size_t)N_NODES * DIM * sizeof(float);   // 51.2 MB
    float* lo      = (float*)(ws);
    float* hi      = (float*)(ws + FILT);
    float* wsrc    = (float*)(ws + 2 * FILT);
    float* wdst    = (float*)(ws + 2 * FILT + 262144);
    float* partial = (float*)(ws + 2 * FILT + 2 * 262144);

    float* out = (float*)d_out;   // out[0] = reg, out[1..E] = gate

    iir_kernel<<<(N_NODES + 255) / 256, 256, 0, stream>>>(
        node_emb, b_hi, a_hi, b_lo, a_lo, hi, lo);

    mlp_wmma_kernel<<<N_NODES / 16, 32, 0, stream>>>(
        lo, hi, idx_src_lo, idx_src_hi, w1_src, b1_src, w2_src, b2_src, wsrc);
    mlp_wmma_kernel<<<N_NODES / 16, 32, 0, stream>>>(
        lo, hi, idx_dst_lo, idx_dst_hi, w1_dst, b1_dst, w2_dst, b2_dst, wdst);

    const int EBLOCKS = 1024;
    edge_kernel<<<EBLOCKS, 256, 0, stream>>>(
        wsrc, wdst, src, dst, u_eps, out + 1, partial, E);
    finalize_kernel<<<1, 256, 0, stream>>>(partial, EBLOCKS, out, 1.0f / (float)E);
}